// ZBLPotential_41944650612790
// MI455X (gfx1250) — compile-verified
//
#include <hip/hip_runtime.h>
#include <hip/hip_bf16.h>
#include <math.h>

#define THREADS 256
#define EPT 4  // edges per thread

typedef __attribute__((address_space(1))) int global_int;
typedef __attribute__((address_space(3))) int lds_int;
typedef __attribute__((ext_vector_type(4))) float fvec4;
typedef __attribute__((ext_vector_type(4))) int ivec4;

struct __align__(8) PowRad {
  float p;  // Z^Z_power
  float r;  // covalent radius
};

__global__ __launch_bounds__(THREADS) void zbl_edges(
    const int* __restrict__ z,
    const float* __restrict__ dist,
    const int* __restrict__ eidx,        // [2, E] flat: sender at [e], receiver at [E+e]
    const float* __restrict__ a_factor,  // scalar
    const float* __restrict__ Z_power,   // scalar
    const float* __restrict__ coefs,     // [4]
    const float* __restrict__ exps,      // [4]
    const float* __restrict__ radii,     // [95]
    float* __restrict__ out,             // [n_nodes]
    int n_edges) {
  __shared__ PowRad s_lut[96];  // packed {Z^zp, radius} per element: one b64 LDS load

  const int tid = threadIdx.x;

  // ---- Issue CDNA5 async global->LDS copy of covalent radii into .y lanes ----
#if __has_builtin(__builtin_amdgcn_global_load_async_to_lds_b32)
  if (tid < 95) {
    __builtin_amdgcn_global_load_async_to_lds_b32(
        (global_int*)(radii + tid), (lds_int*)(&s_lut[tid].r),
        /*offset=*/0, /*cpol=*/0);
  }
#else
  if (tid < 95) s_lut[tid].r = radii[tid];
#endif

  // ---- Overlap: build Z^Z_power LUT while the async copy is in flight ----
  const float zp = Z_power[0];
  if (tid < 95) s_lut[tid].p = (tid == 0) ? 0.0f : __expf(zp * __logf((float)tid));

#if __has_builtin(__builtin_amdgcn_s_wait_asynccnt)
  __builtin_amdgcn_s_wait_asynccnt(0);
#else
  asm volatile("s_wait_asynccnt 0" ::: "memory");
#endif
  __syncthreads();

  const float inv_a = 1.0f / (a_factor[0] * 0.529f);
  const float c0 = coefs[0], c1 = coefs[1], c2 = coefs[2], c3 = coefs[3];
  const float e0 = exps[0], e1 = exps[1], e2 = exps[2], e3 = exps[3];

  auto do_edge = [&](int snd, int rcv, float d) {
    const int Zu = z[snd];
    const int Zv = z[rcv];
    const PowRad lu = s_lut[Zu];  // ds_load_b64
    const PowRad lv = s_lut[Zv];  // ds_load_b64
    const float t = d * __builtin_amdgcn_rcpf(lu.r + lv.r);
    if (t < 1.0f) {  // envelope (and whole edge) is zero otherwise: skip exps + atomic
      const float x = d * (lu.p + lv.p) * inv_a;
      const float phi = c0 * __expf(-e0 * x) + c1 * __expf(-e1 * x) +
                        c2 * __expf(-e2 * x) + c3 * __expf(-e3 * x);
      const float t2 = t * t;
      const float t6 = t2 * t2 * t2;
      // 1 - 28 t^6 + 48 t^7 - 21 t^8 = 1 - t^6 * (28 - 48 t + 21 t^2)
      const float env = 1.0f - t6 * fmaf(t, fmaf(21.0f, t, -48.0f), 28.0f);
      // 0.5 * 14.3996 = 7.1998
      const float contrib =
          7.1998f * (float)(Zu * Zv) * phi * env * __builtin_amdgcn_rcpf(d);
      __hip_atomic_fetch_add(&out[rcv], contrib, __ATOMIC_RELAXED,
                             __HIP_MEMORY_SCOPE_AGENT);
    }
  };

  const int base = (blockIdx.x * THREADS + tid) * EPT;
  if (((n_edges & 3) == 0) && (base + EPT <= n_edges)) {
    // 128-bit non-temporal streaming loads: read-once data, keep L2 for atomics/z
    const fvec4 d4 = __builtin_nontemporal_load((const fvec4*)(dist + base));
    const ivec4 s4 = __builtin_nontemporal_load((const ivec4*)(eidx + base));
    const ivec4 r4 = __builtin_nontemporal_load((const ivec4*)(eidx + n_edges + base));
    do_edge(s4.x, r4.x, d4.x);
    do_edge(s4.y, r4.y, d4.y);
    do_edge(s4.z, r4.z, d4.z);
    do_edge(s4.w, r4.w, d4.w);
  } else {
    for (int e = base; e < n_edges && e < base + EPT; ++e) {
      do_edge(eidx[e], eidx[n_edges + e], dist[e]);
    }
  }
}

__global__ __launch_bounds__(THREADS) void zbl_zero(float* __restrict__ out, int n) {
  int i = blockIdx.x * blockDim.x + threadIdx.x;
  if (i < n) out[i] = 0.0f;
}

extern "C" void kernel_launch(void* const* d_in, const int* in_sizes, int n_in,
                              void* d_out, int out_size, void* d_ws, size_t ws_size,
                              hipStream_t stream) {
  const int* z = (const int*)d_in[0];
  const float* dist = (const float*)d_in[1];
  const int* eidx = (const int*)d_in[2];
  const float* afac = (const float*)d_in[3];
  const float* zpow = (const float*)d_in[4];
  const float* coefs = (const float*)d_in[5];
  const float* sexps = (const float*)d_in[6];
  const float* radii = (const float*)d_in[7];
  float* out = (float*)d_out;

  const int n_nodes = in_sizes[0];
  const int n_edges = in_sizes[1];

  zbl_zero<<<(n_nodes + THREADS - 1) / THREADS, THREADS, 0, stream>>>(out, n_nodes);

  const int threads_total = (n_edges + EPT - 1) / EPT;
  const int blocks = (threads_total + THREADS - 1) / THREADS;
  zbl_edges<<<blocks, THREADS, 0, stream>>>(z, dist, eidx, afac, zpow, coefs,
                                            sexps, radii, out, n_edges);
}